// RadialShuffle_85366769975702
// MI455X (gfx1250) — compile-verified
//
#include <hip/hip_runtime.h>
#include <stdint.h>

#define H_ 224
#define W_ 224
#define HW (H_*W_)
#define NUM_ROWS 159
#define NUM_COLS 1024
#define BATCH 128
#define CENTER 112

__device__ __forceinline__ int ring_of(int x, int y) {
    int dx = x - CENTER, dy = y - CENTER;
    float r = sqrtf((float)(dx * dx + dy * dy));
    return (int)(r + 0.5f);   // safe: sqrt(int) is never within ~8e-4 of a half-integer
}

// ---- Setup kernel 1: per-(ring,row) histogram ------------------------------
__global__ void rs_hist(int* __restrict__ rowcnt) {
    int t = blockIdx.x * blockDim.x + threadIdx.x;
    if (t >= HW) return;
    int y = t / W_, x = t - y * W_;
    int d = ring_of(x, y);
    atomicAdd(&rowcnt[d * H_ + y], 1);
}

// ---- Setup kernel 2: per-ring exclusive prefix over rows -------------------
__global__ void rs_scan(const int* __restrict__ rowcnt,
                        int* __restrict__ rowoff, int* __restrict__ cnt) {
    int d = blockIdx.x * blockDim.x + threadIdx.x;
    if (d >= NUM_ROWS) return;
    int run = 0;
    for (int y = 0; y < H_; ++y) {
        rowoff[d * H_ + y] = run;
        run += rowcnt[d * H_ + y];
    }
    cnt[d] = run;
}

// ---- Setup kernel 3: pixmap[d][ax2] = flat pixel index ---------------------
__global__ void rs_pixmap(const int* __restrict__ rowoff, int* __restrict__ pixmap) {
    int t = blockIdx.x * blockDim.x + threadIdx.x;
    if (t >= HW) return;
    int y = t / W_, x = t - y * W_;
    int d = ring_of(x, y);
    int a = rowoff[d * H_ + y];
    for (int xx = 0; xx < x; ++xx)
        a += (ring_of(xx, y) == d) ? 1 : 0;   // row-major rank within ring & row
    pixmap[d * NUM_COLS + a] = t;
}

// ---- Main kernel: one block per (batch, ring) ------------------------------
__launch_bounds__(256)
__global__ void rs_main(const float* __restrict__ img,
                        const float* __restrict__ mask,
                        const float* __restrict__ perm,
                        const int*   __restrict__ pixmap,
                        const int*   __restrict__ cnt,
                        float* __restrict__ out) {
    __shared__ __align__(16) float s_key[NUM_COLS];
    __shared__ int   s_pix[NUM_COLS];
    __shared__ float s_val[NUM_COLS];
    __shared__ int   s_flag[NUM_COLS];
    __shared__ int   s_scan[256];
    __shared__ __align__(16) unsigned long long c_ord[NUM_COLS];
    __shared__ float c_val[NUM_COLS];

    const int tid = threadIdx.x;
    const int d = blockIdx.x % NUM_ROWS;
    const int b = blockIdx.x / NUM_ROWS;
    const int n = cnt[d];
    const size_t ibase = (size_t)b * HW;

    // CDNA5 async path: stage the contiguous 4KB key row into LDS (ASYNCcnt),
    // overlapped with the scattered image/mask gather below.
    {
        const float* keyrow = perm + ((size_t)b * NUM_ROWS + d) * NUM_COLS;
        unsigned voff = (unsigned)tid * 16u;                      // 256 lanes x 16B = 4KB
        unsigned ldsk = (unsigned)(uintptr_t)(&s_key[0]) + voff;  // low 32 bits = LDS offset
        asm volatile("global_load_async_to_lds_b128 %0, %1, %2"
                     :: "v"(ldsk), "v"(voff), "s"(keyrow) : "memory");
    }

    // Phase 1: gather pixel indices/values/mask; unselected pixels pass through.
    const int* pixrow = pixmap + (size_t)d * NUM_COLS;
    for (int a = tid; a < NUM_COLS; a += 256) {
        int f = 0, sel = 0; float v = 0.0f;
        if (a < n) {
            f = pixrow[a];
            v = img[ibase + f];
            float m = mask[ibase + f];
            sel = (m < 0.5f) ? 1 : 0;
            if (!sel) out[ibase + f] = v;
        }
        s_pix[a]  = f;
        s_val[a]  = v;
        s_flag[a] = sel;
    }

    asm volatile("s_wait_asynccnt 0x0" ::: "memory");   // key row resident in LDS
    __syncthreads();

    // Phase 2: block-wide prefix scan -> compact selected elements in ascending
    // column order. Composite 64-bit order key: (key_bits << 10) | column.
    // Keys are uniform floats in [0,1) (non-negative, no NaN), so IEEE bit
    // patterns are order-isomorphic; low 10 bits reproduce JAX's stable
    // tie-break by column and double as the destination-column lookup.
    const int base4 = tid * 4;
    int f0 = s_flag[base4],     f1 = s_flag[base4 + 1];
    int f2 = s_flag[base4 + 2], f3 = s_flag[base4 + 3];
    int sum = f0 + f1 + f2 + f3;
    s_scan[tid] = sum;
    __syncthreads();
    for (int off = 1; off < 256; off <<= 1) {
        int v   = s_scan[tid];
        int add = (tid >= off) ? s_scan[tid - off] : 0;
        __syncthreads();
        s_scan[tid] = v + add;
        __syncthreads();
    }
    const int K = s_scan[255];
    int j = s_scan[tid] - sum;  // exclusive base for this thread's 4 slots
    #pragma unroll
    for (int q4 = 0; q4 < 4; ++q4) {
        int a = base4 + q4;
        if (s_flag[a]) {
            unsigned kb = __float_as_uint(s_key[a]);
            c_ord[j] = ((unsigned long long)kb << 10) | (unsigned)a;
            c_val[j] = s_val[a];
            ++j;
        }
    }
    __syncthreads();

    // Phase 3: rank = #{i : ord_i < ord_q} (strict; ords unique), then scatter
    // value q to the rank-th selected column (= low 10 bits of c_ord[rank]).
    for (int q = tid; q < K; q += 256) {
        const unsigned long long oq = c_ord[q];
        int r = 0, i = 0;
        for (; i + 4 <= K; i += 4) {
            ulonglong2 o01 = *(const ulonglong2*)&c_ord[i];      // ds_load_b128
            ulonglong2 o23 = *(const ulonglong2*)&c_ord[i + 2];  // ds_load_b128
            r += (o01.x < oq);
            r += (o01.y < oq);
            r += (o23.x < oq);
            r += (o23.y < oq);
        }
        for (; i < K; ++i)
            r += (c_ord[i] < oq);
        int destcol = (int)(c_ord[r] & (unsigned)(NUM_COLS - 1));
        out[ibase + s_pix[destcol]] = c_val[q];
    }
}

extern "C" void kernel_launch(void* const* d_in, const int* in_sizes, int n_in,
                              void* d_out, int out_size, void* d_ws, size_t ws_size,
                              hipStream_t stream) {
    (void)in_sizes; (void)n_in; (void)out_size; (void)ws_size;
    const float* img  = (const float*)d_in[0];
    const float* mask = (const float*)d_in[1];
    const float* perm = (const float*)d_in[2];
    float* out = (float*)d_out;

    int* ws     = (int*)d_ws;
    int* rowcnt = ws;                                   // 159*224 ints
    int* rowoff = ws + NUM_ROWS * H_;                   // 159*224 ints
    int* cnt    = ws + 2 * NUM_ROWS * H_;               // 159 ints
    int* pixmap = ws + 2 * NUM_ROWS * H_ + 256;         // 159*1024 ints  (~937KB total)

    hipMemsetAsync(rowcnt, 0, NUM_ROWS * H_ * sizeof(int), stream);
    rs_hist  <<<(HW + 255) / 256, 256, 0, stream>>>(rowcnt);
    rs_scan  <<<(NUM_ROWS + 63) / 64, 64, 0, stream>>>(rowcnt, rowoff, cnt);
    rs_pixmap<<<(HW + 255) / 256, 256, 0, stream>>>(rowoff, pixmap);
    rs_main  <<<BATCH * NUM_ROWS, 256, 0, stream>>>(img, mask, perm, pixmap, cnt, out);
}